// QuantumKernel_65481071398694
// MI455X (gfx1250) — compile-verified
//
#include <hip/hip_runtime.h>

typedef float v2f __attribute__((ext_vector_type(2)));
typedef float v8f __attribute__((ext_vector_type(8)));

#define QC_NOPS 20

// ---------------------------------------------------------------------------
// Kernel 1: build the fixed 16x16 complex unitary of the theta-dependent
// middle layer. Thread j (j<16) simulates the circuit on basis state e_j,
// producing column j of U. Stored row-major: Ure[n*16+k] = Re(U[n][k]).
// Wire w <-> state-index bit (3-w). One-shot, 16 threads.
// ---------------------------------------------------------------------------
__global__ void qc_build_unitary(const float* __restrict__ theta,
                                 float* __restrict__ Ure,
                                 float* __restrict__ Uim) {
  const int j = threadIdx.x;
  if (j >= 16) return;
  float sr[16], si[16];
#pragma unroll
  for (int n = 0; n < 16; ++n) { sr[n] = (n == j) ? 1.0f : 0.0f; si[n] = 0.0f; }

#pragma unroll
  for (int i = 0; i < QC_NOPS; ++i) {
    const int kind = i % 4;
    if (kind == 3) {
      // CNOT on adjacent wires (c, c+1): control bit (3-c), target bit (2-c)
      const int c  = (i / 4) % 3;
      const int cb = 3 - c, tb = 2 - c;
#pragma unroll
      for (int n = 0; n < 16; ++n) {
        if (((n >> cb) & 1) && !((n >> tb) & 1)) {
          const int p = n | (1 << tb);
          float tr = sr[n], ti = si[n];
          sr[n] = sr[p]; si[n] = si[p];
          sr[p] = tr;    si[p] = ti;
        }
      }
    } else {
      const int w = (7 * i + 3) % 4;
      const int b = 3 - w;
      float s, c;
      __sincosf(theta[i] * 0.5f, &s, &c);   // |arg| ~ N(0,0.5): native is exact enough
#pragma unroll
      for (int n = 0; n < 16; ++n) {
        if (!((n >> b) & 1)) {
          const int p = n | (1 << b);
          const float a0r = sr[n], a0i = si[n], a1r = sr[p], a1i = si[p];
          float n0r, n0i, n1r, n1i;
          if (kind == 0) {            // RX: [[c, -i s], [-i s, c]]
            n0r =  c * a0r + s * a1i;  n0i =  c * a0i - s * a1r;
            n1r =  s * a0i + c * a1r;  n1i = -s * a0r + c * a1i;
          } else if (kind == 1) {     // RY: [[c, -s], [s, c]]
            n0r =  c * a0r - s * a1r;  n0i =  c * a0i - s * a1i;
            n1r =  s * a0r + c * a1r;  n1i =  s * a0i + c * a1i;
          } else {                    // RZ: diag(e^{-i t}, e^{+i t}), t=theta/2
            n0r =  c * a0r + s * a0i;  n0i =  c * a0i - s * a0r;
            n1r =  c * a1r - s * a1i;  n1i =  c * a1i + s * a1r;
          }
          sr[n] = n0r; si[n] = n0i; sr[p] = n1r; si[p] = n1i;
        }
      }
    }
  }
#pragma unroll
  for (int n = 0; n < 16; ++n) {
    Ure[n * 16 + j] = sr[n];   // column j of U
    Uim[n * 16 + j] = si[n];
  }
}

// ---------------------------------------------------------------------------
// Kernel 2: one wave32 per tile of 16 batch elements (grid-stride, ~4 tiles
// per wave to amortize the invariant U preload).
//   A-operand (16x4, f32): rows = state index n, K = encoder basis index
//   B-operand (4x16, f32): cols = batch element, values = product state a_k
//   D = U_re * a  and  D = U_im * a   via 2x4 chained v_wmma_f32_16x16x4_f32
// D layout: lane L holds batch m=L%16, state rows n = v + 8*(L/16), v=0..7.
// Decoder Ry(-y): wires 1..3 rotate in-register, wire 0 via shfl_xor(16).
// ---------------------------------------------------------------------------
__global__ void qc_wmma_main(const float* __restrict__ x,
                             const float* __restrict__ y,
                             const float* __restrict__ Ure,
                             const float* __restrict__ Uim,
                             float* __restrict__ out,
                             int B, int ntiles) {
  const int lane   = threadIdx.x & 31;
  const int waveId = (int)((blockIdx.x * blockDim.x + threadIdx.x) >> 5);
  const int nwaves = (int)((gridDim.x * blockDim.x) >> 5);
  const int half   = lane >> 4;   // K sub-column / state-row half
  const int mrow   = lane & 15;   // A: row n of U;  B/D: batch column

  // Preload U operands (invariant over tiles). k = 4t + 2*half (+0,+1).
  v2f Ar[4], Ai[4];
#pragma unroll
  for (int t = 0; t < 4; ++t) {
    const int k = 4 * t + 2 * half;
    Ar[t] = *(const v2f*)(Ure + mrow * 16 + k);
    Ai[t] = *(const v2f*)(Uim + mrow * 16 + k);
  }

  for (int tile = waveId; tile < ntiles; tile += nwaves) {
    const int m  = tile * 16 + mrow;
    const int mc = (m < B) ? m : (B - 1);        // clamp for safe loads

    // ---- encoder: product-state amplitudes from x -------------------------
    const float4 xv = ((const float4*)x)[mc];
    float s0, c0, s1, c1, s2, c2, s3, c3;
    __sincosf(0.5f * xv.x, &s0, &c0);
    __sincosf(0.5f * xv.y, &s1, &c1);
    __sincosf(0.5f * xv.z, &s2, &c2);
    __sincosf(0.5f * xv.w, &s3, &c3);
    // a_k = ph[k>>2] * pl[k&3]; this lane (B-op col m) needs k&3 = {2h, 2h+1}
    // -> pl pair = (half? s2 : c2) * {c3, s3}: single select, no dynamic index.
    const float ph0 = c0 * c1, ph1 = c0 * s1, ph2 = s0 * c1, ph3 = s0 * s1;
    const float q2  = half ? s2 : c2;
    const float plA = q2 * c3, plB = q2 * s3;

    const v2f Bv0 = {ph0 * plA, ph0 * plB};
    const v2f Bv1 = {ph1 * plA, ph1 * plB};
    const v2f Bv2 = {ph2 * plA, ph2 * plB};
    const v2f Bv3 = {ph3 * plA, ph3 * plB};

    // ---- s = U * a : 8x v_wmma_f32_16x16x4_f32 (Re and Im chains) ---------
    v8f dre = {}; v8f dim = {};
    dre = __builtin_amdgcn_wmma_f32_16x16x4_f32(false, Ar[0], false, Bv0, (short)0, dre, false, false);
    dim = __builtin_amdgcn_wmma_f32_16x16x4_f32(false, Ai[0], false, Bv0, (short)0, dim, false, false);
    dre = __builtin_amdgcn_wmma_f32_16x16x4_f32(false, Ar[1], false, Bv1, (short)0, dre, false, false);
    dim = __builtin_amdgcn_wmma_f32_16x16x4_f32(false, Ai[1], false, Bv1, (short)0, dim, false, false);
    dre = __builtin_amdgcn_wmma_f32_16x16x4_f32(false, Ar[2], false, Bv2, (short)0, dre, false, false);
    dim = __builtin_amdgcn_wmma_f32_16x16x4_f32(false, Ai[2], false, Bv2, (short)0, dim, false, false);
    dre = __builtin_amdgcn_wmma_f32_16x16x4_f32(false, Ar[3], false, Bv3, (short)0, dre, false, false);
    dim = __builtin_amdgcn_wmma_f32_16x16x4_f32(false, Ai[3], false, Bv3, (short)0, dim, false, false);

    // ---- decoder Ry(-y_w) = [[c, s], [-s, c]] -----------------------------
    const float4 yv = ((const float4*)y)[mc];
    float sy0, cy0, sy1, cy1, sy2, cy2, sy3, cy3;
    __sincosf(0.5f * yv.x, &sy0, &cy0);
    __sincosf(0.5f * yv.y, &sy1, &cy1);
    __sincosf(0.5f * yv.z, &sy2, &cy2);
    __sincosf(0.5f * yv.w, &sy3, &cy3);

    float re[8], im[8];
#pragma unroll
    for (int v = 0; v < 8; ++v) { re[v] = dre[v]; im[v] = dim[v]; }

    auto rot = [&](int i0, int i1, float c, float s) {
      const float r0 = re[i0], q0 = im[i0], r1 = re[i1], q1 = im[i1];
      re[i0] =  c * r0 + s * r1;  im[i0] =  c * q0 + s * q1;
      re[i1] = -s * r0 + c * r1;  im[i1] = -s * q0 + c * q1;
    };
    // wire 1 -> bit2 of v
    rot(0, 4, cy1, sy1); rot(1, 5, cy1, sy1); rot(2, 6, cy1, sy1); rot(3, 7, cy1, sy1);
    // wire 2 -> bit1 of v
    rot(0, 2, cy2, sy2); rot(1, 3, cy2, sy2); rot(4, 6, cy2, sy2); rot(5, 7, cy2, sy2);
    // wire 3 -> bit0 of v
    rot(0, 1, cy3, sy3); rot(2, 3, cy3, sy3); rot(4, 5, cy3, sy3); rot(6, 7, cy3, sy3);

    // wire 0 -> bit3 (the lane-half split): exchange with lane^16
    const float sgn = half ? -sy0 : sy0;
#pragma unroll
    for (int v = 0; v < 8; ++v) {
      const float pr = __shfl_xor(re[v], 16, 32);
      const float pi = __shfl_xor(im[v], 16, 32);
      re[v] = cy0 * re[v] + sgn * pr;
      im[v] = cy0 * im[v] + sgn * pi;
    }

    // ---- measurement: Z on wire 0, then abs -------------------------------
    float p = 0.0f;
#pragma unroll
    for (int v = 0; v < 8; ++v) p += re[v] * re[v] + im[v] * im[v];
    const float pp = __shfl_xor(p, 16, 32);   // other half's probability mass
    if (half == 0 && m < B) out[m] = fabsf(p - pp);
  }
}

// ---------------------------------------------------------------------------
extern "C" void kernel_launch(void* const* d_in, const int* in_sizes, int n_in,
                              void* d_out, int out_size, void* d_ws, size_t ws_size,
                              hipStream_t stream) {
  const float* x     = (const float*)d_in[0];
  const float* y     = (const float*)d_in[1];
  const float* theta = (const float*)d_in[2];
  float* out = (float*)d_out;

  float* Ure = (float*)d_ws;        // 256 floats
  float* Uim = Ure + 256;           // 256 floats

  const int B      = in_sizes[0] / 4;       // batch
  const int ntiles = (B + 15) / 16;

  qc_build_unitary<<<1, 32, 0, stream>>>(theta, Ure, Uim);

  const int threads  = 256;                 // 8 waves/block
  const int tilesPerWave = 4;               // amortize U preload
  int blocks = (ntiles + 8 * tilesPerWave - 1) / (8 * tilesPerWave);
  if (blocks < 1) blocks = 1;
  qc_wmma_main<<<blocks, threads, 0, stream>>>(x, y, Ure, Uim, out, B, ntiles);
}